// FloodTraversalSpatial_11682311045643
// MI455X (gfx1250) — compile-verified
//
#include <hip/hip_runtime.h>
#include <hip/hip_bf16.h>

typedef __attribute__((ext_vector_type(2))) float v2f;
typedef __attribute__((ext_vector_type(8))) float v8f;
typedef __attribute__((ext_vector_type(4))) int   v4i;

#define NPTS 8192
#define NB   4
#define KNN  30
#define BIGF 1.0e9f

// ---------------------------------------------------------------------------
// Async global->LDS staging (CDNA5 ASYNCcnt path), with synchronous fallback.
// Builtin signature (from hipcc diagnostic): (v4i global*, v4i shared*, Ii, Ii)
// ---------------------------------------------------------------------------
#if defined(__has_builtin)
#if __has_builtin(__builtin_amdgcn_global_load_async_to_lds_b128) && \
    __has_builtin(__builtin_amdgcn_s_wait_asynccnt)
#define HAVE_ASYNC_LDS 1
#endif
#endif

typedef __attribute__((address_space(1))) v4i v4i_as1;
typedef __attribute__((address_space(3))) v4i v4i_as3;

#ifdef HAVE_ASYNC_LDS
__device__ __forceinline__ void async_copy16(const float* g, float* l) {
    __builtin_amdgcn_global_load_async_to_lds_b128((v4i_as1*)g, (v4i_as3*)l, 0, 0);
}
#define WAIT_ASYNC(n) __builtin_amdgcn_s_wait_asynccnt(n)
#else
__device__ __forceinline__ void async_copy16(const float* g, float* l) {
    const float4 v = *(const float4*)g;
    *(float4*)l = v;
}
#define WAIT_ASYNC(n) ((void)0)
#endif

// ---------------------------------------------------------------------------
// Kernel 1: build WMMA operand arrays + deterministic uniform z init
//   PA[n] = (x, y, z, 1)
//   PB[n] = (-2x, -2y, -2z, |x|^2 + BIG*(C<=0))
// ---------------------------------------------------------------------------
__global__ void prep_kernel(const float* __restrict__ X, const int* __restrict__ C,
                            float* __restrict__ PA, float* __restrict__ PB,
                            float* __restrict__ z0, int BN) {
    int n = blockIdx.x * blockDim.x + threadIdx.x;
    if (n >= BN) return;
    float x = X[(size_t)n * 3 + 0];
    float y = X[(size_t)n * 3 + 1];
    float z = X[(size_t)n * 3 + 2];
    PA[(size_t)n * 4 + 0] = x;
    PA[(size_t)n * 4 + 1] = y;
    PA[(size_t)n * 4 + 2] = z;
    PA[(size_t)n * 4 + 3] = 1.0f;
    float pen = (C[n] > 0) ? 0.0f : BIGF;
    PB[(size_t)n * 4 + 0] = -2.0f * x;
    PB[(size_t)n * 4 + 1] = -2.0f * y;
    PB[(size_t)n * 4 + 2] = -2.0f * z;
    PB[(size_t)n * 4 + 3] = x * x + y * y + z * z + pen;
    // deterministic hash-based uniform in [0,1)
    unsigned u = (unsigned)n * 0x9E3779B9u + 42u;
    u ^= u >> 16; u *= 0x85EBCA6Bu; u ^= u >> 13; u *= 0xC2B2AE35u; u ^= u >> 16;
    z0[n] = (float)(u >> 8) * (1.0f / 16777216.0f);
}

// ---------------------------------------------------------------------------
// Kernel 2: WMMA-tiled pairwise distances + streaming per-row top-30.
// One wave32 per 16-row band. 128 iterations x 64 columns (4 WMMAs each).
// B tiles double-buffered in LDS via async global->LDS copies (ASYNCcnt).
// All 32 lanes maintain per-(row, column-half) top-30 lists; merged at end.
// ---------------------------------------------------------------------------
__global__ __launch_bounds__(32) void knn_kernel(const float* __restrict__ PA,
                                                 const float* __restrict__ PB,
                                                 const int* __restrict__ C,
                                                 int* __restrict__ edgeIdx,
                                                 float* __restrict__ maskW,
                                                 float* __restrict__ denom) {
    __shared__ float tileD[16][65];       // padded: bank-conflict-free scans
    __shared__ float knnD[32][33];        // 32 partial lists (row, col-half)
    __shared__ int   knnI[32][33];
    __shared__ float stage[2][64 * 4];    // double-buffered B staging (64 pts)

    const int b    = blockIdx.y;
    const int i0   = blockIdx.x * 16;
    const int lane = threadIdx.x;
    const int m    = lane & 15;
    const int hi   = lane >> 4;
    const int k0   = hi * 2;              // K components this lane holds

    for (int s = 0; s < 32; ++s) { knnD[lane][s] = 3.0e38f; knnI[lane][s] = 0; }

    // A fragment: row i0+m, components (k0, k0+1). 8-byte aligned (k0 in {0,2}).
    const float* pa = PA + ((size_t)b * NPTS + i0 + m) * 4 + k0;
    v2f a; a.x = pa[0]; a.y = pa[1];

    const float* pbbase = PB + (size_t)b * NPTS * 4;
    const int NT = NPTS / 64;             // 128 stages of 64 columns

    // prologue: stage 0 -> buffer 0 (each lane copies two 16B points)
    async_copy16(pbbase + (size_t)(0 + lane) * 4,  &stage[0][lane * 4]);
    async_copy16(pbbase + (size_t)(32 + lane) * 4, &stage[0][(32 + lane) * 4]);

    for (int it = 0; it < NT; ++it) {
        const int j0  = it * 64;
        const int buf = it & 1;
        if (it + 1 < NT) {
            const int jn = j0 + 64;
            async_copy16(pbbase + (size_t)(jn + lane) * 4,      &stage[buf ^ 1][lane * 4]);
            async_copy16(pbbase + (size_t)(jn + 32 + lane) * 4, &stage[buf ^ 1][(32 + lane) * 4]);
            WAIT_ASYNC(2);    // current stage's 2 copies complete
        } else {
            WAIT_ASYNC(0);
        }
        __syncthreads();      // previous scan done before tileD overwritten

#pragma unroll
        for (int t = 0; t < 4; ++t) {
            // B fragment from LDS: col t*16+m, components (k0, k0+1)
            const float* sp = &stage[buf][(t * 16 + m) * 4 + k0];
            v2f bb; bb.x = sp[0]; bb.y = sp[1];

            v8f c = {};
            // D = A(16x4) x B(4x16); row-constant |x_i|^2 omitted (rank-invariant)
            c = __builtin_amdgcn_wmma_f32_16x16x4_f32(
                    /*neg_a=*/false, a, /*neg_b=*/false, bb,
                    /*c_mod=*/(short)0, c, /*reuse_a=*/false, /*reuse_b=*/false);

            const int colg = j0 + t * 16 + m;
#pragma unroll
            for (int v = 0; v < 8; ++v) {
                const int row = v + hi * 8;       // C/D layout: VGPR v -> row
                float d = c[v];
                if (colg == i0 + row) d += BIGF;  // self exclusion
                tileD[row][t * 16 + m] = d;
            }
        }
        __syncthreads();

        // all 32 lanes scan: lane L owns row L&15, columns [(L>>4)*32, +32)
        {
            const int r  = m;
            const int cb = hi * 32;
            float cur = knnD[lane][KNN - 1];
            for (int cc = 0; cc < 32; ++cc) {
                const float d = tileD[r][cb + cc];
                if (d < cur) {
                    int p = KNN - 1;
                    while (p > 0 && knnD[lane][p - 1] > d) {
                        knnD[lane][p] = knnD[lane][p - 1];
                        knnI[lane][p] = knnI[lane][p - 1];
                        --p;
                    }
                    knnD[lane][p] = d;
                    knnI[lane][p] = j0 + cb + cc;
                    cur = knnD[lane][KNN - 1];
                }
            }
        }
    }
    __syncthreads();

    // merge the two sorted partial lists per row, emit edges + masks + denom
    if (lane < 16) {
        const int r = lane;
        const size_t base = (size_t)b * NPTS + (i0 + r);
        const float mi = (C[base] > 0) ? 1.0f : 0.0f;
        int ia = 0, ib = 0;
        float s = 0.0f;
        for (int k = 0; k < KNN; ++k) {
            const float da = knnD[r][ia];
            const float db = knnD[r + 16][ib];
            int j;
            if (da <= db) { j = knnI[r][ia]; ++ia; }
            else          { j = knnI[r + 16][ib]; ++ib; }
            const float mj = (C[(size_t)b * NPTS + j] > 0) ? 1.0f : 0.0f;
            const float mm = mi * mj;
            edgeIdx[base * 32 + k] = j;
            maskW[base * 32 + k]   = mm;
            s += mm;
        }
        denom[base] = s + 1e-5f;
    }
}

// ---------------------------------------------------------------------------
// Kernel 3: one masked-average smoothing step (alpha == 1.0)
// ---------------------------------------------------------------------------
__global__ void smooth_kernel(const float* __restrict__ zin, float* __restrict__ zout,
                              const int* __restrict__ edgeIdx, const float* __restrict__ maskW,
                              const float* __restrict__ denom, int BN) {
    int n = blockIdx.x * blockDim.x + threadIdx.x;
    if (n >= BN) return;
    const int b = n >> 13;                 // N = 8192
    const float* zb = zin + (size_t)b * NPTS;
    const size_t rb = (size_t)n * 32;
    float s = 0.0f;
#pragma unroll
    for (int k = 0; k < KNN; ++k)
        s += maskW[rb + k] * zb[edgeIdx[rb + k]];
    zout[n] = s / denom[n];
}

// ---------------------------------------------------------------------------
// Kernel 4: pack sort keys: ordered-uint32(z + priority) << 32 | index.
// Low-bits index gives stable ascending argsort tie-breaking.
// ---------------------------------------------------------------------------
__global__ void pack_kernel(const float* __restrict__ z, const float* __restrict__ prio,
                            unsigned long long* __restrict__ sortbuf, int BN) {
    int n = blockIdx.x * blockDim.x + threadIdx.x;
    if (n >= BN) return;
    float v = z[n] + prio[n];
    unsigned u = __float_as_uint(v);
    u = (u & 0x80000000u) ? ~u : (u | 0x80000000u);
    sortbuf[n] = ((unsigned long long)u << 32) | (unsigned)(n & (NPTS - 1));
}

// ---------------------------------------------------------------------------
// Kernel 5: one bitonic compare-exchange stage over each batch row of 8192
// ---------------------------------------------------------------------------
__global__ void bitonic_kernel(unsigned long long* __restrict__ data, int k, int jj) {
    int i = blockIdx.x * blockDim.x + threadIdx.x;   // index within batch row
    int b = blockIdx.y;
    int l = i ^ jj;
    if (l > i) {
        unsigned long long* d = data + (size_t)b * NPTS;
        unsigned long long a0 = d[i];
        unsigned long long a1 = d[l];
        bool asc  = ((i & k) == 0);
        bool swap = asc ? (a0 > a1) : (a0 < a1);
        if (swap) { d[i] = a1; d[l] = a0; }
    }
}

// ---------------------------------------------------------------------------
// Kernel 6: emit permutation indices
// ---------------------------------------------------------------------------
__global__ void out_kernel(const unsigned long long* __restrict__ sortbuf,
                           int* __restrict__ out, int BN) {
    int n = blockIdx.x * blockDim.x + threadIdx.x;
    if (n >= BN) return;
    out[n] = (int)(sortbuf[n] & 0xFFFFFFFFull);
}

// ---------------------------------------------------------------------------
extern "C" void kernel_launch(void* const* d_in, const int* in_sizes, int n_in,
                              void* d_out, int out_size, void* d_ws, size_t ws_size,
                              hipStream_t stream) {
    const int BN = NB * NPTS;

    const float* X    = (const float*)d_in[0];
    const int*   C    = (const int*)d_in[1];
    const float* prio = (const float*)d_in[2];

    char* ws = (char*)d_ws;
    size_t off = 0;
    float* PA    = (float*)(ws + off); off += (size_t)BN * 4 * sizeof(float);        // 512KB
    float* PB    = (float*)(ws + off); off += (size_t)BN * 4 * sizeof(float);        // 512KB
    int*   eIdx  = (int*)(ws + off);   off += (size_t)BN * 32 * sizeof(int);         // 4MB
    float* mW    = (float*)(ws + off); off += (size_t)BN * 32 * sizeof(float);       // 4MB
    float* denom = (float*)(ws + off); off += (size_t)BN * sizeof(float);            // 128KB
    float* z0    = (float*)(ws + off); off += (size_t)BN * sizeof(float);
    float* z1    = (float*)(ws + off); off += (size_t)BN * sizeof(float);
    unsigned long long* sortbuf = (unsigned long long*)(ws + off);

    prep_kernel<<<(BN + 255) / 256, 256, 0, stream>>>(X, C, PA, PB, z0, BN);

    dim3 gknn(NPTS / 16, NB);
    knn_kernel<<<gknn, 32, 0, stream>>>(PA, PB, C, eIdx, mW, denom);

    float* zi = z0;
    float* zo = z1;
    for (int s = 0; s < 5; ++s) {
        smooth_kernel<<<(BN + 255) / 256, 256, 0, stream>>>(zi, zo, eIdx, mW, denom, BN);
        float* t = zi; zi = zo; zo = t;
    }

    pack_kernel<<<(BN + 255) / 256, 256, 0, stream>>>(zi, prio, sortbuf, BN);

    dim3 gsort(NPTS / 256, NB);
    for (int k = 2; k <= NPTS; k <<= 1)
        for (int jj = k >> 1; jj > 0; jj >>= 1)
            bitonic_kernel<<<gsort, 256, 0, stream>>>(sortbuf, k, jj);

    out_kernel<<<(BN + 255) / 256, 256, 0, stream>>>(sortbuf, (int*)d_out, BN);
}